// NeighborExpansionLayer_28372553957846
// MI455X (gfx1250) — compile-verified
//
#include <hip/hip_runtime.h>
#include <hip/hip_bf16.h>

// NeighborExpansionLayer: out[b,h,w, t*8+c] = in[b,h, w+t-4, c] (zero padded), t in 0..8
// Pure data movement: 64 MB read + 604 MB written => HBM-store-bound (~29 us at 23.3 TB/s).
// Strategy: per-(b,h)-row workgroup; async-stage the padded row into LDS (CDNA5
// GLOBAL_LOAD_ASYNC_TO_LDS_B128 + s_wait_asynccnt), then emit the output row as
// coalesced non-temporal b128 stores. Output floats for position w are the
// contiguous LDS window [w*8, w*8+72): chunk i -> w=i/18, j=i%18, src lds[w*8+j*4].

typedef __attribute__((ext_vector_type(4))) float f4;
typedef int v4i __attribute__((vector_size(16)));  // matches builtin param type

#define NB_B 8
#define NB_H 512
#define NB_W 512
#define NB_C 8
#define NB_R 4
#define ROW_IN  (NB_W * NB_C)                   // 4096 floats per input row
#define ROW_OUT (NB_W * NB_C * (2 * NB_R + 1))  // 36864 floats per output row
#define PAD_F   (NB_R * NB_C)                   // 32 floats zero pad each side
#define LDS_F   (ROW_IN + 2 * PAD_F)            // 4160 floats = 16640 B LDS
#define THREADS 256
#define CHUNKS_IN  (ROW_IN / 4)                 // 1024 x 16B chunks
#define CHUNKS_OUT (ROW_OUT / 4)                // 9216 x 16B chunks

__global__ __launch_bounds__(THREADS)
void neighbor_expand_9tap(const float* __restrict__ in, float* __restrict__ out) {
  __shared__ __align__(16) float lds[LDS_F];

  const int tid = threadIdx.x;
  const int row = blockIdx.x;  // flat b*H + h
  const float* inRow  = in  + (size_t)row * ROW_IN;
  float*       outRow = out + (size_t)row * ROW_OUT;

  // Zero the left/right pad so tap reads are branch-free.
  if (tid < PAD_F) {
    lds[tid] = 0.0f;
    lds[PAD_F + ROW_IN + tid] = 0.0f;
  }

#if defined(__gfx1250__) && __has_builtin(__builtin_amdgcn_global_load_async_to_lds_b128)
  // CDNA5 async copy: global -> LDS without a VGPR round trip; tracked by ASYNCcnt.
  #pragma unroll
  for (int it = 0; it < CHUNKS_IN / THREADS; ++it) {
    const int q = it * THREADS + tid;  // 16B chunk index within the row
    __builtin_amdgcn_global_load_async_to_lds_b128(
        (v4i*)(inRow + q * 4),
        (v4i*)&lds[PAD_F + q * 4],
        /*offset=*/0, /*cpol=*/0);
  }
  #if __has_builtin(__builtin_amdgcn_s_wait_asynccnt)
  __builtin_amdgcn_s_wait_asynccnt(0);
  #else
  asm volatile("s_wait_asynccnt 0" ::: "memory");
  #endif
#else
  // Fallback: vector load + ds_store_b128.
  #pragma unroll
  for (int it = 0; it < CHUNKS_IN / THREADS; ++it) {
    const int q = it * THREADS + tid;
    *(f4*)&lds[PAD_F + q * 4] = *(const f4*)(inRow + q * 4);
  }
#endif

  __syncthreads();

  // 9216 16B output chunks per row; consecutive lanes -> consecutive 16B chunks
  // (512B contiguous per wave store). Source window is contiguous in LDS:
  // out chunk (w, j) <- lds[w*8 + j*4 .. +3].
  #pragma unroll 6
  for (int it = 0; it < CHUNKS_OUT / THREADS; ++it) {
    const int i = it * THREADS + tid;
    const int w = i / 18;
    const int j = i - w * 18;
    const f4 v = *(const f4*)&lds[w * 8 + j * 4];
    // Streaming 604MB write, never re-read: keep it out of L2 (TH=NT).
    __builtin_nontemporal_store(v, (f4*)(outRow + (size_t)i * 4));
  }
}

extern "C" void kernel_launch(void* const* d_in, const int* in_sizes, int n_in,
                              void* d_out, int out_size, void* d_ws, size_t ws_size,
                              hipStream_t stream) {
  (void)in_sizes; (void)n_in; (void)d_ws; (void)ws_size; (void)out_size;
  const float* in = (const float*)d_in[0];  // f32 (8,512,512,8)
  float* out = (float*)d_out;               // f32 (8,512,512,72)
  // d_in[1] = radius (4), d_in[2] = dimension (2): compile-time constants here.
  neighbor_expand_9tap<<<dim3(NB_B * NB_H), dim3(THREADS), 0, stream>>>(in, out);
}